// FastFPnet_21689584845493
// MI455X (gfx1250) — compile-verified
//
#include <hip/hip_runtime.h>
#include <math.h>

// ---------------- problem constants ----------------
#define NS    256
#define BS    7
#define UU    8
#define NR    4
#define NT    8
#define DD    2
#define ITERS 3
#define HID   200
#define CIN   32          // 2*NT*D
#define LDH   208         // HID padded to 13*16
#define MROWS (NS*BS*UU)  // 14336 = 896*16
#define P_CONST   100.0f
#define NOISE_C   1e-07f

typedef float v2f __attribute__((ext_vector_type(2)));
typedef float v8f __attribute__((ext_vector_type(8)));

struct cplx { float re, im; };

__device__ __forceinline__ cplx cadd(cplx a, cplx b){ return {a.re+b.re, a.im+b.im}; }
__device__ __forceinline__ cplx csub(cplx a, cplx b){ return {a.re-b.re, a.im-b.im}; }
__device__ __forceinline__ cplx cmul(cplx a, cplx b){ return {a.re*b.re - a.im*b.im, a.re*b.im + a.im*b.re}; }
// a * conj(b)
__device__ __forceinline__ cplx cmulc(cplx a, cplx b){ return {a.re*b.re + a.im*b.im, a.im*b.re - a.re*b.im}; }
__device__ __forceinline__ cplx cmadd(cplx acc, cplx a, cplx b){
    acc.re += a.re*b.re - a.im*b.im; acc.im += a.re*b.im + a.im*b.re; return acc; }
__device__ __forceinline__ cplx cmaddc(cplx acc, cplx a, cplx b){ // acc += a*conj(b)
    acc.re += a.re*b.re + a.im*b.im; acc.im += a.im*b.re - a.re*b.im; return acc; }
__device__ __forceinline__ cplx cinv(cplx a){
    float id = 1.0f/(a.re*a.re + a.im*a.im); return {a.re*id, -a.im*id}; }

// Gauss-Jordan solve of 4x4 complex system with NRHS right-hand sides.
// A is Hermitian PD (noise*I + PSD) -> no pivoting needed. Destroys A; B <- solution.
template<int NRHS>
__device__ void gj4(cplx A[4][4], cplx B[4][NRHS]) {
    for (int c = 0; c < 4; ++c) {
        cplx inv = cinv(A[c][c]);
        for (int cc = 0; cc < 4; ++cc) A[c][cc] = cmul(A[c][cc], inv);
        for (int q = 0; q < NRHS; ++q) B[c][q] = cmul(B[c][q], inv);
        for (int rr = 0; rr < 4; ++rr) {
            if (rr == c) continue;
            cplx f = A[rr][c];
            for (int cc = 0; cc < 4; ++cc) A[rr][cc] = csub(A[rr][cc], cmul(f, A[c][cc]));
            for (int q = 0; q < NRHS; ++q)  B[rr][q]  = csub(B[rr][q],  cmul(f, B[c][q]));
        }
    }
}

// ---------------- K0: initial power normalization ----------------
// one block per (n,k): 128 threads == U*NT*D elements
__global__ __launch_bounds__(128) void k_prescale(const float* __restrict__ Vre,
                                                  const float* __restrict__ Vim,
                                                  cplx* __restrict__ Vc)
{
    int bid = blockIdx.x;                 // n*BS + k
    int tid = threadIdx.x;                // u*16 + t*2 + d
    size_t base = (size_t)bid * (UU*NT*DD);
    float re = Vre[base + tid], im = Vim[base + tid];
    __shared__ float red[128];
    red[tid] = re*re + im*im;
    __syncthreads();
    for (int off = 64; off > 0; off >>= 1) {
        if (tid < off) red[tid] += red[tid + off];
        __syncthreads();
    }
    float scale = sqrtf(P_CONST / red[0]);     // unconditional at init
    Vc[base + tid] = { re*scale, im*scale };
}

// ---------------- K1: per-n Usum, Y = Usum^-1 HV2, IG = I + Gamma ----------------
__global__ __launch_bounds__(256) void k_y_ig(const float* __restrict__ Hre,
                                              const float* __restrict__ Him,
                                              const cplx*  __restrict__ Vc,
                                              cplx* __restrict__ gY,
                                              cplx* __restrict__ gIG)
{
    int n = blockIdx.x, tid = threadIdx.x;
    __shared__ cplx sHV[BS][UU][UU][NR][DD];   // for current k: [l][i][j][r][d]  (28.7 KB)
    __shared__ cplx sUs[BS][UU][NR][NR];       // Usum[l][i][r][s]                (7 KB)
    __shared__ cplx sHV2[BS][UU][NR][DD];      // HV2[l][i][r][d]                 (3.5 KB)

    // init Usum = NOISE * I
    for (int t = tid; t < BS*UU*NR*NR; t += 256) {
        int s = t & 3, r = (t >> 2) & 3, i = (t >> 4) & 7, l = t >> 7;
        sUs[l][i][r][s] = { (r == s) ? NOISE_C : 0.0f, 0.0f };
    }

    for (int k = 0; k < BS; ++k) {
        __syncthreads();
        // HV[l][i][j][r][d] = sum_t H[n,k,l,i,r,t] * V[n,k,j,t,d]
        for (int task = tid; task < BS*UU*UU; task += 256) {
            int j = task & 7, i = (task >> 3) & 7, l = task >> 6;
            size_t hb = ((((size_t)n*BS + k)*BS + l)*UU + i) * (NR*NT);
            const cplx* v = Vc + (((size_t)n*BS + k)*UU + j) * (NT*DD);
            for (int r = 0; r < NR; ++r)
                for (int d = 0; d < DD; ++d) {
                    cplx acc = {0.f, 0.f};
                    for (int t = 0; t < NT; ++t) {
                        cplx h = { Hre[hb + r*NT + t], Him[hb + r*NT + t] };
                        acc = cmadd(acc, h, v[t*DD + d]);
                    }
                    sHV[l][i][j][r][d] = acc;
                }
        }
        __syncthreads();
        // Usum += sum_{j,d} HV[..r..] * conj(HV[..s..])
        for (int task = tid; task < BS*UU*NR*NR; task += 256) {
            int s = task & 3, r = (task >> 2) & 3, i = (task >> 4) & 7, l = task >> 7;
            cplx acc = sUs[l][i][r][s];
            for (int j = 0; j < UU; ++j)
                for (int d = 0; d < DD; ++d)
                    acc = cmaddc(acc, sHV[l][i][j][r][d], sHV[l][i][j][s][d]);
            sUs[l][i][r][s] = acc;
        }
        // capture diagonal HV2 for l==k, j==i
        if (tid < UU*NR*DD) {
            int d = tid & 1, r = (tid >> 1) & 3, i = tid >> 3;
            sHV2[k][i][r][d] = sHV[k][i][i][r][d];
        }
    }
    __syncthreads();

    // per (l,i): solve Y = Usum^-1 HV2 ; F = Usum - hv2 hv2^H ; Gamma = hv2^H F^-1 hv2
    if (tid < BS*UU) {
        int i = tid & 7, l = tid >> 3;
        cplx Am[4][4], Fm[4][4], R[4][2], R2[4][2], hv2[4][2];
        for (int r = 0; r < 4; ++r) {
            for (int s = 0; s < 4; ++s) { Am[r][s] = sUs[l][i][r][s]; Fm[r][s] = Am[r][s]; }
            for (int d = 0; d < 2; ++d) { hv2[r][d] = sHV2[l][i][r][d]; R[r][d] = hv2[r][d]; R2[r][d] = hv2[r][d]; }
        }
        gj4<2>(Am, R);                                  // R = Y
        cplx* y = gY + (((size_t)n*BS + l)*UU + i) * (NR*DD);
        for (int r = 0; r < 4; ++r)
            for (int d = 0; d < 2; ++d) y[r*DD + d] = R[r][d];

        for (int r = 0; r < 4; ++r)
            for (int s = 0; s < 4; ++s) {
                cplx td = cmulc(hv2[r][0], hv2[s][0]);
                td = cmaddc(td, hv2[r][1], hv2[s][1]);
                Fm[r][s] = csub(Fm[r][s], td);
            }
        gj4<2>(Fm, R2);                                 // R2 = F^-1 hv2
        cplx* ig = gIG + (((size_t)n*BS + l)*UU + i) * (DD*DD);
        for (int d = 0; d < 2; ++d)
            for (int e = 0; e < 2; ++e) {
                cplx g = {0.f, 0.f};
                for (int r = 0; r < 4; ++r) g = cmaddc(g, R2[r][e], hv2[r][d]); // conj(hv2)*R2
                if (d == e) g.re += 1.0f;
                ig[d*DD + e] = g;
            }
    }
}

// ---------------- K2: per-(n,l) A, L, Lam, Z2 + MLP input assembly ----------------
__global__ __launch_bounds__(128) void k_z2(const float* __restrict__ Hre,
                                            const float* __restrict__ Him,
                                            const cplx*  __restrict__ Vc,
                                            const cplx*  __restrict__ gY,
                                            const cplx*  __restrict__ gIG,
                                            const float* __restrict__ w,
                                            cplx*  __restrict__ gZ2,
                                            float* __restrict__ act_re,
                                            float* __restrict__ act_im)
{
    int bid = blockIdx.x, tid = threadIdx.x;
    int n = bid / BS, l = bid % BS;
    __shared__ cplx sA[BS][UU][NT][DD];    // A[m][u][t][d]  (7.2 KB)
    __shared__ cplx sC[BS][UU][NT][DD];    // C[m][u][s][d]  (7.2 KB)
    __shared__ cplx sL[UU][NT][NT];        // L[u][t][s]     (4 KB)

    // A[m,u,t,d] = sum_r conj(H[n,l,m,u,r,t]) * Y[n,m,u,r,d]
    for (int task = tid; task < BS*UU*NT*DD; task += 128) {
        int d = task & 1, t = (task >> 1) & 7, u = (task >> 4) & 7, m = task >> 7;
        size_t hb = ((((size_t)n*BS + l)*BS + m)*UU + u) * (NR*NT);
        const cplx* y = gY + (((size_t)n*BS + m)*UU + u) * (NR*DD);
        cplx acc = {0.f, 0.f};
        for (int r = 0; r < NR; ++r) {
            cplx h = { Hre[hb + r*NT + t], Him[hb + r*NT + t] };
            acc = cmaddc(acc, y[r*DD + d], h);   // y * conj(h)
        }
        sA[m][u][t][d] = acc;
    }
    __syncthreads();
    // C[m,u,s,d] = sum_e IG[n,m,u,d,e] * conj(A[m,u,s,e])
    for (int task = tid; task < BS*UU*NT*DD; task += 128) {
        int d = task & 1, s = (task >> 1) & 7, u = (task >> 4) & 7, m = task >> 7;
        const cplx* ig = gIG + (((size_t)n*BS + m)*UU + u) * (DD*DD);
        cplx acc = {0.f, 0.f};
        for (int e = 0; e < DD; ++e) acc = cmaddc(acc, ig[d*DD + e], sA[m][u][s][e]);
        sC[m][u][s][d] = acc;
    }
    __syncthreads();
    // L[u,t,s] = w[u] * sum_{m,d} A[m,u,t,d] * C[m,u,s,d]
    for (int task = tid; task < UU*NT*NT; task += 128) {
        int s = task & 7, t = (task >> 3) & 7, u = task >> 6;
        cplx acc = {0.f, 0.f};
        for (int m = 0; m < BS; ++m)
            for (int d = 0; d < DD; ++d)
                acc = cmadd(acc, sA[m][u][t][d], sC[m][u][s][d]);
        float wu = w[u];
        sL[u][t][s] = { acc.re*wu, acc.im*wu };
    }
    __syncthreads();
    // Z2[u,t,dd] = w[u]*Lam[u,t,dd] - sum_s L[u,t,s]*Z[u,s,dd]
    for (int task = tid; task < UU*NT*DD; task += 128) {
        int dd = task & 1, t = (task >> 1) & 7, u = task >> 4;
        const cplx* ig = gIG + (((size_t)n*BS + l)*UU + u) * (DD*DD);
        cplx lam = {0.f, 0.f};
        for (int d = 0; d < DD; ++d) lam = cmadd(lam, sA[l][u][t][d], ig[d*DD + dd]);
        float wu = w[u];
        lam.re *= wu; lam.im *= wu;
        const cplx* z = Vc + (((size_t)n*BS + l)*UU + u) * (NT*DD);
        cplx acc = lam;
        for (int s = 0; s < NT; ++s) acc = csub(acc, cmul(sL[u][t][s], z[s*DD + dd]));
        size_t row = ((size_t)n*BS + l)*UU + u;
        int col = t*DD + dd;
        gZ2[row*(NT*DD) + col] = acc;
        cplx zv = z[col];
        act_re[row*CIN + col]      = zv.re;  act_im[row*CIN + col]      = zv.im;
        act_re[row*CIN + 16 + col] = acc.re; act_im[row*CIN + 16 + col] = acc.im;
    }
}

// ---------------- K3: complex GEMM via V_WMMA_F32_16X16X4_F32 ----------------
// out[r][o] = crelu( sum_k x[r][k]*W[o][k] + b[o] )   (crelu optional)
// One wave per 16x16 output tile; 4 real WMMAs per K-chunk of 4.
// Branchless inner loop: out-of-range output columns (N..ldo-1, last tile only)
// read weight row 0 (always in-bounds) and are multiplied by a 0 mask.
__global__ __launch_bounds__(32) void k_cgemm(const float* __restrict__ Xre,
                                              const float* __restrict__ Xim,
                                              int ldx, int K,
                                              const float* __restrict__ Wre,
                                              const float* __restrict__ Wim,
                                              const float* __restrict__ bre,
                                              const float* __restrict__ bim,
                                              int N,
                                              float* __restrict__ Ore,
                                              float* __restrict__ Oim,
                                              int ldo, int relu)
{
    int rt = blockIdx.x, ct = blockIdx.y;
    int lane = threadIdx.x;
#if defined(__HIP_DEVICE_COMPILE__) && __has_builtin(__builtin_amdgcn_wmma_f32_16x16x4_f32)
    int mrow = rt*16 + (lane & 15);        // A: lanes 0-15 M=lane, lanes 16-31 M=lane-16
    int half = lane >> 4;                  // K sub-column pair select
    int ocol = ct*16 + (lane & 15);        // B: N striped across lane halves
    bool inb = (ocol < N);
    int ocolc  = inb ? ocol : 0;           // clamped, always in-bounds
    float msk  = inb ? 1.0f : 0.0f;        // zero contribution of pad lanes
    const float* xrp = Xre + (size_t)mrow*ldx + half*2;
    const float* xip = Xim + (size_t)mrow*ldx + half*2;
    const float* wrp = Wre + (size_t)ocolc*K + half*2;
    const float* wip = Wim + (size_t)ocolc*K + half*2;
    v8f accR = {};
    v8f accI = {};
    for (int k0 = 0; k0 < K; k0 += 4) {
        __builtin_prefetch(xrp + 16, 0, 3);    // global_prefetch_b8 (near-cache)
        v2f aR = { xrp[0], xrp[1] };
        v2f aI = { xip[0], xip[1] };
        v2f bR = { wrp[0]*msk, wrp[1]*msk };
        v2f bI = { wip[0]*msk, wip[1]*msk };
        // acc_re += Are*Bre - Aim*Bim ; acc_im += Are*Bim + Aim*Bre
        accR = __builtin_amdgcn_wmma_f32_16x16x4_f32(false, aR, false, bR, (short)0, accR, false, false);
        v2f aIn = { -aI.x, -aI.y };
        accR = __builtin_amdgcn_wmma_f32_16x16x4_f32(false, aIn, false, bI, (short)0, accR, false, false);
        accI = __builtin_amdgcn_wmma_f32_16x16x4_f32(false, aR, false, bI, (short)0, accI, false, false);
        accI = __builtin_amdgcn_wmma_f32_16x16x4_f32(false, aI, false, bR, (short)0, accI, false, false);
        xrp += 4; xip += 4; wrp += 4; wip += 4;
    }
    // D layout: VGPR i -> M = i + 8*(lane>=16); N = lane&15
    int colD  = ct*16 + (lane & 15);
    int rbase = rt*16 + (lane >> 4)*8;
    float br = inb ? bre[ocolc] : 0.f;
    float bi = inb ? bim[ocolc] : 0.f;
    for (int i = 0; i < 8; ++i) {
        float vr = accR[i] + br;
        float vi = accI[i] + bi;
        if (relu) vr = fmaxf(vr, 0.f);
        int r = rbase + i;
        Ore[(size_t)r*ldo + colD] = vr;
        Oim[(size_t)r*ldo + colD] = vi;
    }
#else
    // scalar fallback (also used for host-side stub compilation)
    int colD  = ct*16 + (lane & 15);
    int rbase = rt*16 + (lane >> 4)*8;
    for (int i = 0; i < 8; ++i) {
        int r = rbase + i;
        float ar = 0.f, ai = 0.f;
        if (colD < N) {
            for (int k = 0; k < K; ++k) {
                float xr = Xre[(size_t)r*ldx + k], xi = Xim[(size_t)r*ldx + k];
                float wr = Wre[(size_t)colD*K + k], wi = Wim[(size_t)colD*K + k];
                ar += xr*wr - xi*wi;
                ai += xr*wi + xi*wr;
            }
            ar += bre[colD]; ai += bim[colD];
        }
        if (relu) ar = fmaxf(ar, 0.f);
        Ore[(size_t)r*ldo + colD] = ar;
        Oim[(size_t)r*ldo + colD] = ai;
    }
#endif
}

// ---------------- K4: output head (N=1): step = Re( h . W4 + b4 ) ----------------
__global__ __launch_bounds__(256) void k_head(const float* __restrict__ Hre,
                                              const float* __restrict__ Him,
                                              int ld,
                                              const float* __restrict__ W4re,
                                              const float* __restrict__ W4im,
                                              const float* __restrict__ b4re,
                                              float* __restrict__ step, int mrows)
{
    int row = blockIdx.x * blockDim.x + threadIdx.x;
    if (row >= mrows) return;
    float acc = b4re[0];
    const float* hr = Hre + (size_t)row*ld;
    const float* hi = Him + (size_t)row*ld;
    for (int i = 0; i < HID; ++i)
        acc += hr[i]*W4re[i] - hi[i]*W4im[i];
    step[row] = acc;
}

// ---------------- K5: V update + conditional renormalization ----------------
__global__ __launch_bounds__(128) void k_update(cplx* __restrict__ Vc,
                                                const cplx* __restrict__ gZ2,
                                                const float* __restrict__ step)
{
    int bid = blockIdx.x;            // n*BS + l
    int tid = threadIdx.x;           // u*16 + t*2 + d
    int u = tid >> 4;
    size_t base = (size_t)bid * (UU*NT*DD);
    size_t row  = (size_t)bid * UU + u;
    cplx v  = Vc[base + tid];
    cplx z2 = gZ2[base + tid];
    float s = step[row];
    v.re += s * z2.re;
    v.im += s * z2.im;
    __shared__ float red[128];
    red[tid] = v.re*v.re + v.im*v.im;
    __syncthreads();
    for (int off = 64; off > 0; off >>= 1) {
        if (tid < off) red[tid] += red[tid + off];
        __syncthreads();
    }
    float fro = red[0];
    float scale = (fro > P_CONST) ? sqrtf(P_CONST / fro) : 1.0f;
    Vc[base + tid] = { v.re*scale, v.im*scale };
}

// ---------------- K6: emit stacked (re, im) output ----------------
__global__ void k_out(const cplx* __restrict__ Vc, float* __restrict__ out, int ntot)
{
    int i = blockIdx.x * blockDim.x + threadIdx.x;
    if (i < ntot) { out[2*i] = Vc[i].re; out[2*i + 1] = Vc[i].im; }
}

// ---------------- launcher ----------------
extern "C" void kernel_launch(void* const* d_in, const int* in_sizes, int n_in,
                              void* d_out, int out_size, void* d_ws, size_t ws_size,
                              hipStream_t stream)
{
    (void)in_sizes; (void)n_in; (void)out_size; (void)ws_size;
    const float* Hre = (const float*)d_in[0];
    const float* Him = (const float*)d_in[1];
    const float* Vre = (const float*)d_in[2];
    const float* Vim = (const float*)d_in[3];
    const float* w   = (const float*)d_in[4];
    const float* W1re = (const float*)d_in[5],  *W1im = (const float*)d_in[6];
    const float* b1re = (const float*)d_in[7],  *b1im = (const float*)d_in[8];
    const float* W2re = (const float*)d_in[9],  *W2im = (const float*)d_in[10];
    const float* b2re = (const float*)d_in[11], *b2im = (const float*)d_in[12];
    const float* W3re = (const float*)d_in[13], *W3im = (const float*)d_in[14];
    const float* b3re = (const float*)d_in[15], *b3im = (const float*)d_in[16];
    const float* W4re = (const float*)d_in[17], *W4im = (const float*)d_in[18];
    const float* b4re = (const float*)d_in[19];
    // d_in[20] = b4_im (unused: only the real part of the head feeds step)

    char* ws = (char*)d_ws;
    size_t off = 0;
    auto alloc = [&](size_t bytes) -> void* {
        void* p = ws + off;
        off = (off + bytes + 255) & ~(size_t)255;
        return p;
    };
    const size_t nV = (size_t)NS*BS*UU*NT*DD;             // 229,376 complex
    cplx*  Vc   = (cplx*) alloc(nV * sizeof(cplx));
    cplx*  gY   = (cplx*) alloc((size_t)NS*BS*UU*NR*DD * sizeof(cplx));
    cplx*  gIG  = (cplx*) alloc((size_t)NS*BS*UU*DD*DD * sizeof(cplx));
    cplx*  gZ2  = (cplx*) alloc(nV * sizeof(cplx));
    float* a0re = (float*)alloc((size_t)MROWS*CIN * sizeof(float));
    float* a0im = (float*)alloc((size_t)MROWS*CIN * sizeof(float));
    float* h1re = (float*)alloc((size_t)MROWS*LDH * sizeof(float));
    float* h1im = (float*)alloc((size_t)MROWS*LDH * sizeof(float));
    float* h2re = (float*)alloc((size_t)MROWS*LDH * sizeof(float));
    float* h2im = (float*)alloc((size_t)MROWS*LDH * sizeof(float));
    float* stepb= (float*)alloc((size_t)MROWS * sizeof(float));

    // initial normalization
    k_prescale<<<NS*BS, 128, 0, stream>>>(Vre, Vim, Vc);

    dim3 gmm(MROWS/16, LDH/16);   // 896 x 13 tiles

    for (int it = 0; it < ITERS; ++it) {
        k_y_ig<<<NS, 256, 0, stream>>>(Hre, Him, Vc, gY, gIG);
        k_z2 <<<NS*BS, 128, 0, stream>>>(Hre, Him, Vc, gY, gIG, w, gZ2, a0re, a0im);

        const float* w1r = W1re + (size_t)it*HID*CIN; const float* w1i = W1im + (size_t)it*HID*CIN;
        const float* w2r = W2re + (size_t)it*HID*HID; const float* w2i = W2im + (size_t)it*HID*HID;
        const float* w3r = W3re + (size_t)it*HID*HID; const float* w3i = W3im + (size_t)it*HID*HID;

        k_cgemm<<<gmm, 32, 0, stream>>>(a0re, a0im, CIN, CIN,
                                        w1r, w1i, b1re + (size_t)it*HID, b1im + (size_t)it*HID, HID,
                                        h1re, h1im, LDH, 1);
        k_cgemm<<<gmm, 32, 0, stream>>>(h1re, h1im, LDH, HID,
                                        w2r, w2i, b2re + (size_t)it*HID, b2im + (size_t)it*HID, HID,
                                        h2re, h2im, LDH, 1);
        k_cgemm<<<gmm, 32, 0, stream>>>(h2re, h2im, LDH, HID,
                                        w3r, w3i, b3re + (size_t)it*HID, b3im + (size_t)it*HID, HID,
                                        h1re, h1im, LDH, 1);
        k_head<<<(MROWS + 255)/256, 256, 0, stream>>>(h1re, h1im, LDH,
                                                      W4re + (size_t)it*HID, W4im + (size_t)it*HID,
                                                      b4re + it, stepb, MROWS);
        k_update<<<NS*BS, 128, 0, stream>>>(Vc, gZ2, stepb);
    }

    k_out<<<((int)nV + 255)/256, 256, 0, stream>>>(Vc, (float*)d_out, (int)nV);
}